// NNUE_21680994910623
// MI455X (gfx1250) — compile-verified
//
#include <hip/hip_runtime.h>

typedef __attribute__((ext_vector_type(2))) float v2f;
typedef __attribute__((ext_vector_type(8))) float v8f;

#define NFEAT      768
#define PAD_IDX    768
#define L1W        512      // accumulator width
#define NB         8        // buckets
#define BUCKET_DIV 4
#define SPW        16       // samples per workgroup
#define ROWSTRIDE  1028     // 1024 + 4 floats pad -> conflict-free ds_load_b64

__global__ __launch_bounds__(256)
void nnue_fwd_kernel(const int* __restrict__ stm,
                     const int* __restrict__ nstm,
                     const float* __restrict__ emb,     // [769, 512]
                     const float* __restrict__ bias,    // [512]
                     const float* __restrict__ hw,      // [8, 1024]
                     const float* __restrict__ hb,      // [8]
                     float* __restrict__ out,           // [B]
                     int Btot)
{
    __shared__ float embS[SPW * ROWSTRIDE];   // activated embeddings, 16 x 1024 (padded)
    __shared__ float hwS[16 * ROWSTRIDE];     // hidden_w padded to 16 rows
    __shared__ float outS[16 * 16];           // 16 samples x 16 (8 buckets + pad)
    __shared__ int   bucketS[SPW];

    const int t      = threadIdx.x;
    const int wave   = t >> 5;
    const int lane   = t & 31;
    const int wgBase = blockIdx.x * SPW;
    if (wgBase >= Btot) return;

    // ---- Phase 0: init out tile, stage hidden_w (rows 8..15 = 0) ----
    outS[t] = 0.0f;
    for (int i = t; i < 16 * 1024; i += 256) {
        const int r = i >> 10, c = i & 1023;
        hwS[r * ROWSTRIDE + c] = (r < NB) ? hw[r * 1024 + c] : 0.0f;
    }

    // per-lane bias slice: bias[lane*16 + 0..15] (same for both sides/samples)
    float bia[16];
    #pragma unroll
    for (int i = 0; i < 4; ++i) {
        const float4 b4 = *(const float4*)(bias + lane * 16 + i * 4);
        bia[4 * i + 0] = b4.x; bia[4 * i + 1] = b4.y;
        bia[4 * i + 2] = b4.z; bia[4 * i + 3] = b4.w;
    }

    // ---- Phase 1: each wave gathers+accumulates 2 samples ----
    for (int sub = 0; sub < 2; ++sub) {
        const int m = wave * 2 + sub;            // sample index within workgroup
        const int s = wgBase + m;                // global sample
        const int idx_s = stm[s * 32 + lane];    // one index per lane
        const int idx_n = nstm[s * 32 + lane];

        // material bucket from stm non-pad count
        const unsigned long long bm = __ballot(idx_s != PAD_IDX);
        int bkt = (__popcll(bm) - 2) / BUCKET_DIV;
        bkt = bkt < 0 ? 0 : (bkt > NB - 1 ? NB - 1 : bkt);
        if (lane == 0) bucketS[m] = bkt;

        float accS[16], accN[16];
        #pragma unroll
        for (int i = 0; i < 16; ++i) { accS[i] = 0.0f; accN[i] = 0.0f; }

        #pragma unroll 4
        for (int j = 0; j < 32; ++j) {
            const int ri = __builtin_amdgcn_readlane(idx_s, j);  // uniform -> SGPR base
            if (ri != PAD_IDX) {
                const float* row = emb + (size_t)ri * L1W + lane * 16;
                #pragma unroll
                for (int i = 0; i < 4; ++i) {
                    const float4 r4 = *(const float4*)(row + 4 * i);
                    accS[4 * i + 0] += r4.x; accS[4 * i + 1] += r4.y;
                    accS[4 * i + 2] += r4.z; accS[4 * i + 3] += r4.w;
                }
            }
            const int rn = __builtin_amdgcn_readlane(idx_n, j);
            if (rn != PAD_IDX) {
                const float* row = emb + (size_t)rn * L1W + lane * 16;
                #pragma unroll
                for (int i = 0; i < 4; ++i) {
                    const float4 r4 = *(const float4*)(row + 4 * i);
                    accN[4 * i + 0] += r4.x; accN[4 * i + 1] += r4.y;
                    accN[4 * i + 2] += r4.z; accN[4 * i + 3] += r4.w;
                }
            }
        }

        // activation: clamp(acc + bias, 0, 1)^2, store to LDS [m][0..1023]
        float* dstS = embS + m * ROWSTRIDE + lane * 16;
        float* dstN = dstS + L1W;
        #pragma unroll
        for (int i = 0; i < 16; ++i) {
            float v = fminf(fmaxf(accS[i] + bia[i], 0.0f), 1.0f);
            dstS[i] = v * v;
            float w = fminf(fmaxf(accN[i] + bia[i], 0.0f), 1.0f);
            dstN[i] = w * w;
        }
    }

    __syncthreads();

    // ---- Phase 2: head GEMM  C[16 samples x 16 buckets] += A[16xK] * B[Kx16]
    // fp32 WMMA 16x16x4; each wave owns a K-slab of 128, cross-wave ds_add_f32.
    {
        const int half = lane >> 4;     // 0: K+0/1, 1: K+2/3
        const int n    = lane & 15;     // A: sample row M; B: bucket col N
        const float* ap = embS + n * ROWSTRIDE + 2 * half;
        const float* bp = hwS  + n * ROWSTRIDE + 2 * half;
        const int kbase = wave * 128;

        v8f c = {};
        #pragma unroll 4
        for (int kk = 0; kk < 128; kk += 4) {
            const int k = kbase + kk;
            const v2f a = *(const v2f*)(ap + k);
            const v2f b = *(const v2f*)(bp + k);
            c = __builtin_amdgcn_wmma_f32_16x16x4_f32(
                    /*neg_a=*/false, a, /*neg_b=*/false, b,
                    /*c_mod=*/(short)0, c, /*reuse_a=*/false, /*reuse_b=*/false);
        }

        #pragma unroll
        for (int i = 0; i < 8; ++i)
            atomicAdd(&outS[(i + 8 * half) * 16 + n], c[i]);  // ds_add_f32
    }

    __syncthreads();

    // ---- Phase 3: bucket select + bias, one output per sample ----
    if (t < SPW) {
        const int s = wgBase + t;
        if (s < Btot) {
            const int bkt = bucketS[t];
            out[s] = outS[t * 16 + bkt] + hb[bkt];
        }
    }
}

extern "C" void kernel_launch(void* const* d_in, const int* in_sizes, int n_in,
                              void* d_out, int out_size, void* d_ws, size_t ws_size,
                              hipStream_t stream) {
    const int*   stm  = (const int*)d_in[0];     // [B,32]
    const int*   nstm = (const int*)d_in[1];     // [B,32]
    const float* emb  = (const float*)d_in[2];   // [769,512]
    const float* bias = (const float*)d_in[3];   // [512]
    const float* hw   = (const float*)d_in[4];   // [8,1024]
    const float* hb   = (const float*)d_in[5];   // [8]
    float* out = (float*)d_out;

    const int Btot = in_sizes[0] / 32;
    const int grid = (Btot + SPW - 1) / SPW;
    nnue_fwd_kernel<<<grid, 256, 0, stream>>>(stm, nstm, emb, bias, hw, hb, out, Btot);
}